// GcnSAGE_5609227288771
// MI455X (gfx1250) — compile-verified
//
#include <hip/hip_runtime.h>
#include <math.h>

typedef __attribute__((ext_vector_type(2))) float v2f;
typedef __attribute__((ext_vector_type(8))) float v8f;

#define EPSLN 1e-5f

// D = A(16x4 f32) * B(4x16 f32) + C(16x16 f32), exact fp32 WMMA on gfx1250.
static __device__ __forceinline__ v8f wmma4(v2f a, v2f b, v8f c) {
  return __builtin_amdgcn_wmma_f32_16x16x4_f32(
      /*neg_a=*/false, a, /*neg_b=*/false, b,
      /*c_mod=*/(short)0, c, /*reuse_a=*/false, /*reuse_b=*/false);
}

// ---------------------------------------------------------------------------
// Input projector: x[N,192] = concat_j leakyrelu(LN64(h[:,chunk_j] @ Wj^T + bj))
// One block = 32 node rows (two 16-row WMMA tiles, B fragment reused).
// 8 waves cover the 12 (chunk, 16-col-tile) pairs; wave index scalarized via
// readfirstlane so all loop control is SGPR-uniform (EXEC stays all-1s).
// ---------------------------------------------------------------------------
__global__ __launch_bounds__(256)
void proj_kernel(const float* __restrict__ h,
                 const float* __restrict__ W0, const float* __restrict__ b0,
                 const float* __restrict__ g0, const float* __restrict__ be0,
                 const float* __restrict__ W1, const float* __restrict__ b1,
                 const float* __restrict__ g1, const float* __restrict__ be1,
                 const float* __restrict__ W2, const float* __restrict__ b2,
                 const float* __restrict__ g2, const float* __restrict__ be2,
                 float* __restrict__ x, int N)
{
  __shared__ float lds_out[32 * 192];
  const int tid  = threadIdx.x;
  const int lane = tid & 31;
  const int wave = __builtin_amdgcn_readfirstlane(tid >> 5);  // SGPR
  const int lo = lane & 15, hi = lane >> 4;
  const int rowBase = blockIdx.x * 32;

  for (int p = wave; p < 12; p += 8) {              // uniform loop
    const int chunk = p >> 2, nt = p & 3;           // SGPRs
    const float* Wc = (chunk == 0) ? W0 : (chunk == 1) ? W1 : W2;
    const float* bc = (chunk == 0) ? b0 : (chunk == 1) ? b1 : b2;
    const int Kc    = (chunk == 0) ? 4  : (chunk == 1) ? 300 : 720;
    const int off   = (chunk == 0) ? 0  : (chunk == 1) ? 4   : 304;

    const int col = nt * 16 + lo;
    const float bias = bc[col];
    v8f acc0, acc1;
    #pragma unroll
    for (int r = 0; r < 8; ++r) { acc0[r] = bias; acc1[r] = bias; }

    int row0 = rowBase + lo;      if (row0 >= N) row0 = N - 1;
    int row1 = rowBase + 16 + lo; if (row1 >= N) row1 = N - 1;
    const float* h0 = h  + (long)row0 * 1024 + off + 2 * hi;  // A: lane=row, K striped
    const float* h1 = h  + (long)row1 * 1024 + off + 2 * hi;
    const float* wr = Wc + (long)col  * Kc  + 2 * hi;         // B: lane=col, K striped
    #pragma unroll 4
    for (int k = 0; k < Kc; k += 4) {
      v2f b;  b.x  = wr[k]; b.y  = wr[k + 1];
      v2f a0; a0.x = h0[k]; a0.y = h0[k + 1];
      v2f a1; a1.x = h1[k]; a1.y = h1[k + 1];
      acc0 = wmma4(a0, b, acc0);                    // B reused across row tiles
      acc1 = wmma4(a1, b, acc1);
    }
    // D layout: VGPR r, lane L -> M = r + 8*(L/16), Nc = L%16
    #pragma unroll
    for (int r = 0; r < 8; ++r) {
      lds_out[(r + 8 * hi) * 192 + chunk * 64 + nt * 16 + lo]        = acc0[r];
      lds_out[(16 + r + 8 * hi) * 192 + chunk * 64 + nt * 16 + lo]   = acc1[r];
    }
  }
  __syncthreads();

  // LayerNorm over each 64-wide chunk + LeakyReLU(0.01): 32 rows x 3 chunks
  if (tid < 96) {
    const int row = tid & 31, chunk = tid >> 5;
    const float* gs  = (chunk == 0) ? g0  : (chunk == 1) ? g1  : g2;
    const float* bes = (chunk == 0) ? be0 : (chunk == 1) ? be1 : be2;
    const float* srcp = lds_out + row * 192 + chunk * 64;
    float mu = 0.f;
    for (int j = 0; j < 64; ++j) mu += srcp[j];
    mu *= (1.f / 64.f);
    float var = 0.f;
    for (int j = 0; j < 64; ++j) { float d = srcp[j] - mu; var += d * d; }
    var *= (1.f / 64.f);
    const float rs = rsqrtf(var + EPSLN);
    const int rowG = rowBase + row;
    if (rowG < N) {
      float* dstp = x + (long)rowG * 192 + chunk * 64;
      for (int j = 0; j < 64; ++j) {
        float v = (srcp[j] - mu) * rs * gs[j] + bes[j];
        dstp[j] = (v > 0.f) ? v : 0.01f * v;
      }
    }
  }
}

// ---------------------------------------------------------------------------
// Degree count + 1/deg (inf -> 0)
// ---------------------------------------------------------------------------
__global__ void deg_kernel(const int* __restrict__ dst, float* __restrict__ deg, int E) {
  for (int e = blockIdx.x * blockDim.x + threadIdx.x; e < E; e += gridDim.x * blockDim.x)
    atomicAdd(&deg[dst[e]], 1.0f);
}

__global__ void norm_kernel(float* __restrict__ nrm, const float* __restrict__ deg, int N) {
  int i = blockIdx.x * blockDim.x + threadIdx.x;
  if (i < N) { float d = deg[i]; nrm[i] = (d > 0.f) ? 1.f / d : 0.f; }
}

// ---------------------------------------------------------------------------
// Edge scatter: ah[dst] += x[src] * dist  (norm folded in at GEMM staging)
// F threads per edge; coalesced row gather, global_atomic_add_f32 scatter.
// Prefetch the next edge's source row (global_prefetch_b8) — the gather is
// the HBM-bound part of this model.
// ---------------------------------------------------------------------------
template<int F>
__global__ void edge_agg_kernel(const float* __restrict__ x, const int* __restrict__ src,
                                const int* __restrict__ dst, const float* __restrict__ dist,
                                float* __restrict__ ah, int E)
{
  const int f = threadIdx.x;
  for (int e = blockIdx.x; e < E; e += gridDim.x) {
    const int e2 = e + gridDim.x;
    if (e2 < E)
      __builtin_prefetch(&x[(long)src[e2] * F + f], 0, 1);
    const int s = src[e], d = dst[e];
    const float w = dist[e];
    atomicAdd(&ah[(long)d * F + f], x[(long)s * F + f] * w);
  }
}

// ---------------------------------------------------------------------------
// SAGE layer: y = [x | ah*norm] @ W^T + b, optional LN(NOUT)+ReLU.
// One block = 32 rows (two 16-row tiles, B reused); A staged in LDS;
// wave nt (scalarized) owns output cols [16nt, 16nt+16).
// ---------------------------------------------------------------------------
template<int KIN, int NOUT, bool LNACT>
__global__ __launch_bounds__(256)
void sage_kernel(const float* __restrict__ xin, const float* __restrict__ ah,
                 const float* __restrict__ nrm,
                 const float* __restrict__ W, const float* __restrict__ bvec,
                 const float* __restrict__ g, const float* __restrict__ be,
                 float* __restrict__ y, int N)
{
  constexpr int KA  = 2 * KIN;
  constexpr int NT  = (NOUT + 15) / 16;
  constexpr int NOP = NT * 16;
  __shared__ float lds_a[32 * KA];
  __shared__ float lds_out[32 * NOP];

  const int tid  = threadIdx.x;
  const int lane = tid & 31;
  const int wave = __builtin_amdgcn_readfirstlane(tid >> 5);  // SGPR
  const int lo = lane & 15, hi = lane >> 4;
  const int rowBase = blockIdx.x * 32;

  // Stage concat(x, ah*norm) tile [32 x KA]
  for (int t = tid; t < 32 * KA; t += 256) {
    const int r = t / KA, c = t % KA;
    int row = rowBase + r; if (row >= N) row = N - 1;
    float v;
    if (c < KIN) v = xin[(long)row * KIN + c];
    else         v = ah[(long)row * KIN + (c - KIN)] * nrm[row];
    lds_a[t] = v;
  }
  __syncthreads();

  if (wave < NT) {                                  // uniform branch
    const int nt  = wave;
    const int col = nt * 16 + lo;
    const bool valid = (col < NOUT);
    const float bias = valid ? bvec[col] : 0.f;
    v8f acc0, acc1;
    #pragma unroll
    for (int r = 0; r < 8; ++r) { acc0[r] = bias; acc1[r] = bias; }

    const float* a0p = lds_a + lo * KA + 2 * hi;
    const float* a1p = lds_a + (16 + lo) * KA + 2 * hi;
    const float* wr  = W + (long)(valid ? col : 0) * KA + 2 * hi;
    #pragma unroll 4
    for (int k = 0; k < KA; k += 4) {
      v2f b;
      b.x = valid ? wr[k]     : 0.f;
      b.y = valid ? wr[k + 1] : 0.f;
      v2f a0; a0.x = a0p[k]; a0.y = a0p[k + 1];
      v2f a1; a1.x = a1p[k]; a1.y = a1p[k + 1];
      acc0 = wmma4(a0, b, acc0);
      acc1 = wmma4(a1, b, acc1);
    }
    #pragma unroll
    for (int r = 0; r < 8; ++r) {
      lds_out[(r + 8 * hi) * NOP + nt * 16 + lo]      = acc0[r];
      lds_out[(16 + r + 8 * hi) * NOP + nt * 16 + lo] = acc1[r];
    }
  }
  __syncthreads();

  if constexpr (LNACT) {
    if (tid < 32) {
      const int row = tid;
      float mu = 0.f;
      for (int j = 0; j < NOUT; ++j) mu += lds_out[row * NOP + j];
      mu *= (1.f / (float)NOUT);
      float var = 0.f;
      for (int j = 0; j < NOUT; ++j) { float d = lds_out[row * NOP + j] - mu; var += d * d; }
      var *= (1.f / (float)NOUT);
      const float rs = rsqrtf(var + EPSLN);
      const int rowG = rowBase + row;
      if (rowG < N) {
        for (int j = 0; j < NOUT; ++j) {
          float v = (lds_out[row * NOP + j] - mu) * rs * g[j] + be[j];
          y[(long)rowG * NOUT + j] = fmaxf(v, 0.f);
        }
      }
    }
  } else {
    if (tid < 32 * NOUT) {
      const int row = tid / NOUT, j = tid % NOUT;
      const int rowG = rowBase + row;
      if (rowG < N)
        y[(long)rowG * NOUT + j] = lds_out[row * NOP + j];
    }
  }
}

// ---------------------------------------------------------------------------
extern "C" void kernel_launch(void* const* d_in, const int* in_sizes, int n_in,
                              void* d_out, int out_size, void* d_ws, size_t ws_size,
                              hipStream_t stream)
{
  const float* h    = (const float*)d_in[0];
  const int*   src  = (const int*)  d_in[1];
  const int*   dstE = (const int*)  d_in[2];
  const float* dist = (const float*)d_in[3];
  const float* pW0 = (const float*)d_in[4];  const float* pb0 = (const float*)d_in[5];
  const float* pg0 = (const float*)d_in[6];  const float* pbe0= (const float*)d_in[7];
  const float* pW1 = (const float*)d_in[8];  const float* pb1 = (const float*)d_in[9];
  const float* pg1 = (const float*)d_in[10]; const float* pbe1= (const float*)d_in[11];
  const float* pW2 = (const float*)d_in[12]; const float* pb2 = (const float*)d_in[13];
  const float* pg2 = (const float*)d_in[14]; const float* pbe2= (const float*)d_in[15];
  const float* lW0 = (const float*)d_in[16]; const float* lb0 = (const float*)d_in[17];
  const float* lW1 = (const float*)d_in[18]; const float* lb1 = (const float*)d_in[19];
  const float* lW2 = (const float*)d_in[20]; const float* lb2 = (const float*)d_in[21];
  const float* lg0 = (const float*)d_in[22]; const float* lbe0= (const float*)d_in[23];
  const float* lg1 = (const float*)d_in[24]; const float* lbe1= (const float*)d_in[25];
  (void)n_in; (void)out_size; (void)ws_size;

  const int N = in_sizes[0] / 1024;
  const int E = in_sizes[1];

  float* ws  = (float*)d_ws;
  float* x0  = ws;                       // N*192 projector output
  float* ahb = x0  + (size_t)N * 192;    // N*192 aggregation buffer (reused)
  float* y1  = ahb + (size_t)N * 192;    // N*96
  float* y2  = y1  + (size_t)N * 96;     // N*96
  float* deg = y2  + (size_t)N * 96;     // N
  float* nrm = deg + (size_t)N;          // N

  const int rowBlocks = (N + 31) / 32;

  hipMemsetAsync(deg, 0, sizeof(float) * (size_t)N, stream);
  hipMemsetAsync(ahb, 0, sizeof(float) * (size_t)N * 192, stream);

  proj_kernel<<<rowBlocks, 256, 0, stream>>>(h,
      pW0, pb0, pg0, pbe0, pW1, pb1, pg1, pbe1, pW2, pb2, pg2, pbe2, x0, N);

  deg_kernel<<<4096, 256, 0, stream>>>(dstE, deg, E);
  norm_kernel<<<(N + 255) / 256, 256, 0, stream>>>(nrm, deg, N);

  // Layer 0: 384 -> 96, LN + ReLU
  edge_agg_kernel<192><<<8192, 192, 0, stream>>>(x0, src, dstE, dist, ahb, E);
  sage_kernel<192, 96, true><<<rowBlocks, 256, 0, stream>>>(
      x0, ahb, nrm, lW0, lb0, lg0, lbe0, y1, N);

  // Layer 1: 192 -> 96, LN + ReLU
  hipMemsetAsync(ahb, 0, sizeof(float) * (size_t)N * 96, stream);
  edge_agg_kernel<96><<<8192, 96, 0, stream>>>(y1, src, dstE, dist, ahb, E);
  sage_kernel<96, 96, true><<<rowBlocks, 256, 0, stream>>>(
      y1, ahb, nrm, lW1, lb1, lg1, lbe1, y2, N);

  // Layer 2: 192 -> 5, no LN / act
  hipMemsetAsync(ahb, 0, sizeof(float) * (size_t)N * 96, stream);
  edge_agg_kernel<96><<<8192, 96, 0, stream>>>(y2, src, dstE, dist, ahb, E);
  sage_kernel<96, 5, false><<<rowBlocks, 256, 0, stream>>>(
      y2, ahb, nrm, lW2, lb2, nullptr, nullptr, (float*)d_out, N);
}